// Graph_Generator_55035710931542
// MI455X (gfx1250) — compile-verified
//
#include <hip/hip_runtime.h>
#include <math.h>

#define Bq   16
#define Nn   2048
#define Ll   12
#define Cc   32
#define KTOP 1638                       // int(2048 * 0.8)
#define SCALE 0.17677669529663687f      // 1/sqrt(32)

typedef __attribute__((ext_vector_type(2))) float v2f;
typedef __attribute__((ext_vector_type(8))) float v8f;

// map float -> uint with same total order as float comparison
__device__ __forceinline__ unsigned ordf(float f) {
    unsigned u = __float_as_uint(f);
    return (u & 0x80000000u) ? ~u : (u | 0x80000000u);
}
__device__ __forceinline__ float wredmax(float v) {
#pragma unroll
    for (int o = 16; o > 0; o >>= 1) v = fmaxf(v, __shfl_xor(v, o, 32));
    return v;
}
__device__ __forceinline__ float wredsum(float v) {
#pragma unroll
    for (int o = 16; o > 0; o >>= 1) v += __shfl_xor(v, o, 32);
    return v;
}

// Kernel 1: xsum[b,n,c] = sum_l x[b,n,l,c]
__global__ void xsum_kernel(const float* __restrict__ x, float* __restrict__ xsum) {
    int idx = blockIdx.x * blockDim.x + threadIdx.x;
    if (idx >= Bq * Nn * Cc) return;
    int c  = idx & (Cc - 1);
    int bn = idx >> 5;
    const float* p = x + (size_t)bn * (Ll * Cc) + c;
    float s = 0.f;
#pragma unroll
    for (int l = 0; l < Ll; ++l) s += p[l * Cc];
    xsum[idx] = s;
}

// Kernel 2: fully fused dual-GEMM + softmaxes + top-k mask + final softmax.
// One workgroup = one batch b, 16 output rows. 16 waves.
__launch_bounds__(512)
__global__ void fused_graph_kernel(const float* __restrict__ xsum,
                                   const float* __restrict__ mem,   // (C, N) row-major
                                   const float* __restrict__ fcw,
                                   const float* __restrict__ fcb,
                                   float* __restrict__ out) {
    __shared__ float sAdj[16 * Nn];   // s1 scores -> adj -> masked adj   (128 KB)
    __shared__ float sS2 [16 * Nn];   // s2 scores -> radix histograms    (128 KB)

    const int b    = blockIdx.x >> 7;          // 128 row-tiles per batch
    const int n0   = (blockIdx.x & 127) << 4;
    const int lane = threadIdx.x & 31;
    const int w    = threadIdx.x >> 5;         // wave id 0..15

    const float w0 = fcw[0], w1 = fcw[1], bias = fcb[0];

    // WMMA f32 16x16x4 fragment coordinates (wave32):
    // A: lane -> M = lane%16, half -> K pair base; B mirrored; C/D: VGPR v -> M = v + 8*(lane>=16)
    const int Mrow = lane & 15;
    const int kb   = (lane >> 4) << 1;
    const int rr   = (lane >> 4) << 3;

    // A fragments: xsum rows n0..n0+15 (constant across all column tiles)
    const float* Arow = xsum + ((size_t)b * Nn + n0 + Mrow) * Cc;
    v2f afr[8];
#pragma unroll
    for (int kk = 0; kk < 8; ++kk) {
        afr[kk][0] = Arow[kk * 4 + kb];
        afr[kk][1] = Arow[kk * 4 + kb + 1];
    }

    // ---- dual GEMM over 128 column tiles (8 per wave) ----
    for (int t = w; t < 128; t += 16) {
        const int m0 = t << 4;
        v8f acc1 = {};
        v8f acc2 = {};
        const float* Brow2 = xsum + ((size_t)b * Nn + m0 + Mrow) * Cc; // B col index == Mrow
#pragma unroll
        for (int kk = 0; kk < 8; ++kk) {
            const int c0 = kk * 4 + kb;
            v2f b1, b2;
            b1[0] = mem[(size_t)c0 * Nn + m0 + Mrow];
            b1[1] = mem[(size_t)(c0 + 1) * Nn + m0 + Mrow];
            b2[0] = Brow2[c0];
            b2[1] = Brow2[c0 + 1];
            acc1 = __builtin_amdgcn_wmma_f32_16x16x4_f32(false, afr[kk], false, b1,
                                                         (short)0, acc1, false, false);
            acc2 = __builtin_amdgcn_wmma_f32_16x16x4_f32(false, afr[kk], false, b2,
                                                         (short)0, acc2, false, false);
        }
#pragma unroll
        for (int v = 0; v < 8; ++v) {
            const int r = v + rr;
            sAdj[r * Nn + m0 + Mrow] = fmaxf(acc1[v] * SCALE, 0.f);  // relu(s1*scale)
            sS2 [r * Nn + m0 + Mrow] = fmaxf(acc2[v] * SCALE, 0.f);  // relu(s2*scale)
        }
    }
    __syncthreads();

    // ---- per-row phases: wave w owns row w; float4 sweeps (ds_*_b128) ----
    float*  rowA = &sAdj[w * Nn];
    float*  rowB = &sS2 [w * Nn];
    float4* rA4  = (float4*)rowA;     // Nn/4 = 512 vectors, 16B aligned
    float4* rB4  = (float4*)rowB;

    // softmax stats for a1, a2 (values >= 0 after relu, so max >= 0)
    float mx1 = 0.f, mx2 = 0.f;
    for (int i = lane; i < Nn / 4; i += 32) {
        float4 a = rA4[i], c = rB4[i];
        mx1 = fmaxf(mx1, fmaxf(fmaxf(a.x, a.y), fmaxf(a.z, a.w)));
        mx2 = fmaxf(mx2, fmaxf(fmaxf(c.x, c.y), fmaxf(c.z, c.w)));
    }
    mx1 = wredmax(mx1); mx2 = wredmax(mx2);
    float su1 = 0.f, su2 = 0.f;
    for (int i = lane; i < Nn / 4; i += 32) {
        float4 a = rA4[i], c = rB4[i];
        su1 += __expf(a.x - mx1) + __expf(a.y - mx1) + __expf(a.z - mx1) + __expf(a.w - mx1);
        su2 += __expf(c.x - mx2) + __expf(c.y - mx2) + __expf(c.z - mx2) + __expf(c.w - mx2);
    }
    su1 = wredsum(su1); su2 = wredsum(su2);
    const float i1 = w0 / su1, i2 = w1 / su2;
    for (int i = lane; i < Nn / 4; i += 32) {
        float4 a = rA4[i], c = rB4[i], d;
        d.x = __expf(a.x - mx1) * i1 + __expf(c.x - mx2) * i2 + bias;
        d.y = __expf(a.y - mx1) * i1 + __expf(c.y - mx2) * i2 + bias;
        d.z = __expf(a.z - mx1) * i1 + __expf(c.z - mx2) * i2 + bias;
        d.w = __expf(a.w - mx1) * i1 + __expf(c.w - mx2) * i2 + bias;
        rA4[i] = d;                                   // adj
    }
    __syncthreads();

    // ---- radix select: threshold = KTOP-th largest of adj row ----
    unsigned* hist = (unsigned*)rowB;   // 256 bins in this wave's private stripe
    unsigned prefix = 0;
    int kneed = KTOP;
#pragma unroll 1
    for (int p = 3; p >= 0; --p) {
        for (int i = lane; i < 256; i += 32) hist[i] = 0u;
        __syncthreads();
        const unsigned highmask = (p == 3) ? 0u : (0xFFFFFFFFu << ((p + 1) * 8));
        for (int i = lane; i < Nn / 4; i += 32) {
            float4 a = rA4[i];
            unsigned u0 = ordf(a.x), u1 = ordf(a.y), u2 = ordf(a.z), u3 = ordf(a.w);
            if ((u0 & highmask) == prefix) atomicAdd(&hist[(u0 >> (p * 8)) & 0xFFu], 1u);
            if ((u1 & highmask) == prefix) atomicAdd(&hist[(u1 >> (p * 8)) & 0xFFu], 1u);
            if ((u2 & highmask) == prefix) atomicAdd(&hist[(u2 >> (p * 8)) & 0xFFu], 1u);
            if ((u3 & highmask) == prefix) atomicAdd(&hist[(u3 >> (p * 8)) & 0xFFu], 1u);
        }
        __syncthreads();
        int selBin = 0, cumAbove = 0;
        if (lane == 0) {
            int cum = 0;
            for (int bin = 255; bin >= 0; --bin) {
                int cb = (int)hist[bin];
                if (cum + cb >= kneed) { selBin = bin; cumAbove = cum; break; }
                cum += cb;
            }
        }
        selBin   = __shfl(selBin, 0, 32);
        cumAbove = __shfl(cumAbove, 0, 32);
        kneed  -= cumAbove;
        prefix |= ((unsigned)selBin) << (p * 8);
        __syncthreads();
    }
    const unsigned uT    = prefix;   // exact bit pattern of k-th largest
    const int     keepEq = kneed;    // how many ties (==uT) to keep, lowest index first

    // ---- mask (tie-aware, strict column order) ----
    int   eqseen = 0;
    float mmax   = 0.f;              // masked row contains zeros, so max >= 0
#pragma unroll 1
    for (int j = 0; j < Nn / 32; ++j) {
        const int c = j * 32 + lane;
        float v = rowA[c];
        unsigned u = ordf(v);
        bool eq = (u == uT);
        unsigned bal = (unsigned)__ballot(eq);
        int before = eqseen + __popc(bal & ((1u << lane) - 1u));
        bool keep = (u > uT) || (eq && before < keepEq);
        eqseen += __popc(bal);
        float mv = keep ? v : 0.f;
        rowA[c] = mv;
        mmax = fmaxf(mmax, mv);
    }
    mmax = wredmax(mmax);

    // ---- final softmax over masked row, vectorized ----
    float ssum = 0.f;
    for (int i = lane; i < Nn / 4; i += 32) {
        float4 a = rA4[i];
        ssum += __expf(a.x - mmax) + __expf(a.y - mmax) + __expf(a.z - mmax) + __expf(a.w - mmax);
    }
    ssum = wredsum(ssum);
    const float inv = 1.f / ssum;
    float4* orow4 = (float4*)(out + ((size_t)b * Nn + n0 + w) * Nn);
    for (int i = lane; i < Nn / 4; i += 32) {
        float4 a = rA4[i], r;
        r.x = __expf(a.x - mmax) * inv;
        r.y = __expf(a.y - mmax) * inv;
        r.z = __expf(a.z - mmax) * inv;
        r.w = __expf(a.w - mmax) * inv;
        orow4[i] = r;                 // global_store_b128, coalesced
    }
}

extern "C" void kernel_launch(void* const* d_in, const int* in_sizes, int n_in,
                              void* d_out, int out_size, void* d_ws, size_t ws_size,
                              hipStream_t stream) {
    const float* x    = (const float*)d_in[0];   // (B, N, L, C)
    const float* memp = (const float*)d_in[1];   // (C, N)
    const float* fcw  = (const float*)d_in[2];   // (1, 2)
    const float* fcb  = (const float*)d_in[3];   // (1,)
    float* out = (float*)d_out;                  // (B, N, N)
    float* xs  = (float*)d_ws;                   // B*N*C floats = 4 MB scratch

    const int total = Bq * Nn * Cc;
    xsum_kernel<<<(total + 255) / 256, 256, 0, stream>>>(x, xs);
    fused_graph_kernel<<<Bq * (Nn / 16), 512, 0, stream>>>(xs, memp, fcw, fcb, out);
}